// FocalLoss_35356170780964
// MI455X (gfx1250) — compile-verified
//
#include <hip/hip_runtime.h>

// Problem constants (from setup_inputs): B=4, A=65536, C=80, M=32
#define A_TOTAL 65536
#define BATCH   4
#define NCLS    80
#define NANN    32
#define TPB     256
#define BPI     (A_TOTAL / TPB)     // 256 blocks per image
#define NWAVES  (TPB / 32)
#define FEPS    1.0e-4f

typedef __attribute__((ext_vector_type(2))) float v2f;
typedef __attribute__((ext_vector_type(4))) float v4f;
typedef __attribute__((ext_vector_type(8))) float v8f;

// Wave32 sum via the CDNA5 matrix pipe:
//   A (16x4 f32): lane L (0..15) holds row M=L: {K0=a.x, K1=a.y}; lanes 16..31
//   hold K2/K3 of rows 0..15.  With a.y=0 and B = ones(4x16):
//   D[m][n] = v[m] + v[m+16] for every n.
//   C/D layout: VGPR r, lanes 0..15 -> row r; lanes 16..31 -> row r+8.
//   So sum(d[0..7]) on lanes<16 = rows 0..7, on lanes>=16 = rows 8..15;
//   one shfl_xor(16) gives the full 32-lane sum in every lane.
__device__ __forceinline__ float wave_sum32(float v) {
    v2f a; a.x = v;    a.y = 0.0f;
    v2f b; b.x = 1.0f; b.y = 1.0f;
    v8f c = {};
    c = __builtin_amdgcn_wmma_f32_16x16x4_f32(false, a, false, b,
                                              (short)0, c, false, false);
    float s = ((c[0] + c[1]) + (c[2] + c[3])) + ((c[4] + c[5]) + (c[6] + c[7]));
    s += __shfl_xor(s, 16, 32);
    return s;
}

__device__ __forceinline__ float clip01(float p) {
    return fminf(fmaxf(p, FEPS), 1.0f - FEPS);
}

__global__ __launch_bounds__(TPB)
void focal_main(const float* __restrict__ inst,
                const float* __restrict__ cls,
                const float* __restrict__ regr,
                const float* __restrict__ anch,
                const float* __restrict__ ann,
                float* __restrict__ partials) {
    __shared__ float sann[NANN][8];          // x1,y1,x2,y2,area,cls,flag,pad
    __shared__ float swave[NWAVES][5];

    const int img = blockIdx.x / BPI;
    const int blk = blockIdx.x % BPI;
    const int tid = threadIdx.x;

    if (tid < NANN) {
        const float* ap = ann + ((size_t)img * NANN + tid) * 6;
        float x1 = ap[0], y1 = ap[1], x2 = ap[2], y2 = ap[3];
        sann[tid][0] = x1; sann[tid][1] = y1;
        sann[tid][2] = x2; sann[tid][3] = y2;
        sann[tid][4] = (x2 - x1) * (y2 - y1);   // area
        sann[tid][5] = ap[4];                   // class
        sann[tid][6] = ap[5];                   // flag
    }
    __syncthreads();

    const int a = blk * TPB + tid;

    // anchors: (A,4) = [y1, x1, y2, x2]
    const v4f an = *(const v4f*)(anch + (size_t)a * 4);
    const float ay1 = an[0], ax1 = an[1], ay2 = an[2], ax2 = an[3];
    const float aw = ax2 - ax1, ah = ay2 - ay1;
    const float aarea = aw * ah;

    // ---- IoU max / argmax over 32 annotations (first-max, strict >) ----
    float best = -1.0f; int argm = 0;
    #pragma unroll 4
    for (int j = 0; j < NANN; ++j) {
        const float bx1 = sann[j][0], by1 = sann[j][1];
        const float bx2 = sann[j][2], by2 = sann[j][3];
        float iw = fmaxf(fminf(ax2, bx2) - fmaxf(ax1, bx1), 0.0f);
        float ih = fmaxf(fminf(ay2, by2) - fmaxf(ay1, by1), 0.0f);
        float inter = iw * ih;
        float ua = fmaxf(aarea + sann[j][4] - inter, 1.0e-8f);
        float iou = inter / ua;
        if (iou > best) { best = iou; argm = j; }
    }

    const bool pos      = best >= 0.5f;
    const bool pos_inst = best >= 0.3f;
    const bool neg      = best <  0.4f;
    const float npos  = pos      ? 1.0f : 0.0f;
    const float nposi = pos_inst ? 1.0f : 0.0f;

    // ---- instance focal loss ----
    float x = clip01(inst[(size_t)img * A_TOTAL + a]);
    const float flag = sann[argm][6];
    float instl = 0.0f;
    if (pos_inst) {
        if (flag == 1.0f)      { float u = 1.0f - x; instl = 0.5f * u * u * (-__logf(x)); }
        else if (flag == 0.0f) { instl = 0.5f * x * x * (-__logf(1.0f - x)); }
    }

    // ---- regression smooth-L1 (positive anchors only) ----
    float regl = 0.0f;
    if (pos) {
        const float bx1 = sann[argm][0], by1 = sann[argm][1];
        const float bx2 = sann[argm][2], by2 = sann[argm][3];
        const float acx = ax1 + 0.5f * aw, acy = ay1 + 0.5f * ah;
        const float gww = fmaxf(bx2 - bx1, 1.0f);
        const float ghh = fmaxf(by2 - by1, 1.0f);
        const float gcx = bx1 + 0.5f * (bx2 - bx1);
        const float gcy = by1 + 0.5f * (by2 - by1);
        const float t0 = (gcy - acy) / ah;       // tdy
        const float t1 = (gcx - acx) / aw;       // tdx
        const float t2 = __logf(ghh / ah);       // tdh
        const float t3 = __logf(gww / aw);       // tdw
        const v4f rg = *(const v4f*)(regr + ((size_t)img * A_TOTAL + a) * 4);
        const float tt[4] = { t0, t1, t2, t3 };
        #pragma unroll
        for (int i = 0; i < 4; ++i) {
            float d = fabsf(tt[i] - rg[i]);
            regl += (d <= (1.0f / 9.0f)) ? 4.5f * d * d : d - (0.5f / 9.0f);
        }
    }

    // ---- classification focal loss over 80 classes ----
    // Ignored anchors (0.4 <= iou < 0.5) contribute 0: skip their 320B row.
    float clsl = 0.0f;
    if (pos || neg) {
        const v4f* cp = (const v4f*)(cls + ((size_t)img * A_TOTAL + a) * NCLS);
        #pragma unroll
        for (int q = 0; q < NCLS / 4; ++q) {
            v4f p4 = __builtin_nontemporal_load(cp + q);   // streaming, NT hint
            #pragma unroll
            for (int i = 0; i < 4; ++i) {
                float p = clip01(p4[i]);
                float u = 1.0f - p;
                clsl += 0.75f * p * p * (-__logf(u));       // target==0 branch
            }
        }
        if (pos) {
            int k = (int)sann[argm][5];
            k = min(max(k, 0), NCLS - 1);
            float pk = clip01(cls[((size_t)img * A_TOTAL + a) * NCLS + k]);
            float uk = 1.0f - pk;
            // replace the t==0 term by the t==1 term for the target class
            clsl += 0.25f * uk * uk * (-__logf(pk)) - 0.75f * pk * pk * (-__logf(uk));
        }
    }

    // ---- wave reduction via v_wmma_f32_16x16x4_f32, then block reduce ----
    const float s_cls = wave_sum32(clsl);
    const float s_ins = wave_sum32(instl);
    const float s_reg = wave_sum32(regl);
    const float s_np  = wave_sum32(npos);
    const float s_npi = wave_sum32(nposi);

    const int wave = tid >> 5;
    if ((tid & 31) == 0) {
        swave[wave][0] = s_cls; swave[wave][1] = s_ins; swave[wave][2] = s_reg;
        swave[wave][3] = s_np;  swave[wave][4] = s_npi;
    }
    __syncthreads();
    if (tid < 5) {
        float t = 0.0f;
        #pragma unroll
        for (int w = 0; w < NWAVES; ++w) t += swave[w][tid];
        partials[(size_t)blockIdx.x * 5 + tid] = t;
    }
}

__global__ __launch_bounds__(TPB)
void focal_finalize(const float* __restrict__ partials, float* __restrict__ out) {
    __shared__ float sw[NWAVES][5];
    const int tid = threadIdx.x;
    float accI = 0.0f, accC = 0.0f, accR = 0.0f;

    for (int img = 0; img < BATCH; ++img) {
        const float* p = partials + ((size_t)img * BPI + tid) * 5;
        float q0 = p[0], q1 = p[1], q2 = p[2], q3 = p[3], q4 = p[4];
        #pragma unroll
        for (int off = 16; off >= 1; off >>= 1) {
            q0 += __shfl_xor(q0, off, 32);
            q1 += __shfl_xor(q1, off, 32);
            q2 += __shfl_xor(q2, off, 32);
            q3 += __shfl_xor(q3, off, 32);
            q4 += __shfl_xor(q4, off, 32);
        }
        if ((tid & 31) == 0) {
            int w = tid >> 5;
            sw[w][0] = q0; sw[w][1] = q1; sw[w][2] = q2; sw[w][3] = q3; sw[w][4] = q4;
        }
        __syncthreads();
        if (tid == 0) {
            float cs = 0, is = 0, rs = 0, np = 0, npi = 0;
            #pragma unroll
            for (int w = 0; w < NWAVES; ++w) {
                cs += sw[w][0]; is += sw[w][1]; rs += sw[w][2];
                np += sw[w][3]; npi += sw[w][4];
            }
            accI += is / fmaxf(npi, 1.0f);
            accC += cs / fmaxf(np, 1.0f);
            accR += (np > 0.0f) ? rs / fmaxf(4.0f * np, 1.0f) : 0.0f;
        }
        __syncthreads();
    }
    if (tid == 0) {
        out[0] = accI * (1.0f / BATCH);   // mean instance loss
        out[1] = accC * (1.0f / BATCH);   // mean classification loss
        out[2] = accR * (1.0f / BATCH);   // mean regression loss
    }
}

extern "C" void kernel_launch(void* const* d_in, const int* in_sizes, int n_in,
                              void* d_out, int out_size, void* d_ws, size_t ws_size,
                              hipStream_t stream) {
    const float* inst = (const float*)d_in[0];   // (B, A, 1)
    const float* cls  = (const float*)d_in[1];   // (B, A, 80)
    const float* regr = (const float*)d_in[2];   // (B, A, 4)
    const float* anch = (const float*)d_in[3];   // (1, A, 4)
    const float* ann  = (const float*)d_in[4];   // (B, 32, 6)
    float* out       = (float*)d_out;            // 3 floats
    float* partials  = (float*)d_ws;             // B*BPI*5 floats = 20 KB, fully
                                                 // overwritten each call

    focal_main<<<dim3(BATCH * BPI), dim3(TPB), 0, stream>>>(
        inst, cls, regr, anch, ann, partials);
    focal_finalize<<<dim3(1), dim3(TPB), 0, stream>>>(partials, out);
}